// FundamentalRefinerModule_32358283608460
// MI455X (gfx1250) — compile-verified
//
#include <hip/hip_runtime.h>
#include <math.h>

typedef __attribute__((ext_vector_type(2))) float v2f;
typedef __attribute__((ext_vector_type(8))) float v8f;

#define N_PTS    4096
#define N_MODELS 1024
#define LM_ITERS 20
#define THREADS  128          // 4 waves of 32
#define WAVES    4
#define PTS_PER_WAVE (N_PTS / WAVES)      // 1024
#define TILE     128
#define TILES_PER_WAVE (PTS_PER_WAVE / TILE)  // 8
#define EPSF 1e-12f

struct SharedLM {
    float w[N_PTS];            // per-point weights (16 KB)
    float S[WAVES][TILE * 16]; // per-wave staging of sqrt(w)*[J | r | 0pad] (32 KB)
    float Mred[WAVES][256];    // per-wave 16x16 partial Grammians (4 KB)
    float M[100];              // reduced 10x10 augmented normal matrix
    float F[9];                // current model
    float Fn[9];               // candidate model
    float red[THREADS];        // block reduction buffer
    float lam, cost, maj;
};

__device__ __forceinline__ float block_reduce(float v, float* red) {
    int t = threadIdx.x;
    red[t] = v;
    __syncthreads();
    for (int s = THREADS / 2; s > 0; s >>= 1) {
        if (t < s) red[t] += red[t + s];
        __syncthreads();
    }
    float r = red[0];
    __syncthreads();
    return r;
}

// s-term of oriented epipolar constraint: ((e2 x x2h) . (F x1h))
__device__ __forceinline__ float chir_s(const float* f, float4 p) {
    float x1 = p.x, y1 = p.y, x2 = p.z, y2 = p.w;
    float l0 = f[0]*x1 + f[1]*y1 + f[2];
    float l1 = f[3]*x1 + f[4]*y1 + f[5];
    float l2 = f[6]*x1 + f[7]*y1 + f[8];
    float e20 = f[3]*f[7] - f[6]*f[4];
    float e21 = f[6]*f[1] - f[0]*f[7];
    float e22 = f[0]*f[4] - f[3]*f[1];
    float c0 = e21        - e22 * y2;
    float c1 = e22 * x2   - e20;
    float c2 = e20 * y2   - e21 * x2;
    return c0*l0 + c1*l1 + c2*l2;
}

__device__ __forceinline__ void epi_terms(const float* f, float4 p,
                                          float& e, float& jjt) {
    float x1 = p.x, y1 = p.y, x2 = p.z, y2 = p.w;
    float a0 = f[0]*x1 + f[1]*y1 + f[2];
    float a1 = f[3]*x1 + f[4]*y1 + f[5];
    float b0 = f[0]*x2 + f[3]*y2 + f[6];
    float b1 = f[1]*x2 + f[4]*y2 + f[7];
    float b2 = f[2]*x2 + f[5]*y2 + f[8];
    e   = b0*x1 + b1*y1 + b2;                       // x2^T F x1
    jjt = a0*a0 + a1*a1 + b0*b0 + b1*b1 + 1e-12f;
}

__device__ __forceinline__ float cost_pass(const float* Fsh, const float4* pts4,
                                           const float* w, float* red) {
    float f[9];
#pragma unroll
    for (int i = 0; i < 9; ++i) f[i] = Fsh[i];
    float local = 0.f;
    int t = threadIdx.x;
#pragma unroll 4
    for (int p = 0; p < N_PTS / THREADS; ++p) {
        int n = p * THREADS + t;
        float e, jjt;
        epi_terms(f, pts4[n], e, jjt);
        float r2 = e * e / jjt;
        local += w[n] * log1pf(r2);                 // c2 == 1
    }
    return block_reduce(local, red);
}

__global__ void k_pw(const float* __restrict__ prob,
                     const float* __restrict__ prob_pow,
                     float* __restrict__ pw) {
    int n = blockIdx.x * blockDim.x + threadIdx.x;
    if (n < N_PTS) {
        float p = fmaxf(prob[n], 0.f);
        pw[n] = powf(p, prob_pow[0]);
    }
}

__global__ void __launch_bounds__(THREADS)
k_lm(const float* __restrict__ points, const float* __restrict__ models,
     const float* __restrict__ pw, float* __restrict__ Fout,
     int* __restrict__ counts) {
    __shared__ SharedLM sh;
    const float4* pts4 = (const float4*)points;
    int t = threadIdx.x, lane = t & 31, wv = t >> 5;
    int m = blockIdx.x;

    // --- init: normalized F0 ---
    if (t == 0) {
        float f[9], nrm = 0.f;
#pragma unroll
        for (int i = 0; i < 9; ++i) { f[i] = models[m * 9 + i]; nrm += f[i] * f[i]; }
        float inv = 1.f / (sqrtf(nrm) + EPSF);
#pragma unroll
        for (int i = 0; i < 9; ++i) sh.F[i] = f[i] * inv;
        sh.lam = 1e-3f;
    }
    __syncthreads();

    // --- cheirality: majority sign ---
    {
        float f[9];
#pragma unroll
        for (int i = 0; i < 9; ++i) f[i] = sh.F[i];
        float loc = 0.f;
        for (int p = 0; p < N_PTS / THREADS; ++p) {
            float s = chir_s(f, pts4[p * THREADS + t]);
            loc += (s > 0.f ? 1.f : 0.f) - (s < 0.f ? 1.f : 0.f);
        }
        float tot = block_reduce(loc, sh.red);
        if (t == 0) sh.maj = (tot > 0.f) ? 1.f : ((tot < 0.f) ? -1.f : 1.f);
        __syncthreads();
    }

    // --- weights + nonzero count ---
    {
        float f[9];
#pragma unroll
        for (int i = 0; i < 9; ++i) f[i] = sh.F[i];
        float maj = sh.maj;
        float cnt = 0.f;
        for (int p = 0; p < N_PTS / THREADS; ++p) {
            int n = p * THREADS + t;
            float s = chir_s(f, pts4[n]);
            float wn = (s * maj > 0.f) ? pw[n] : 0.f;
            sh.w[n] = wn;
            cnt += (wn > 0.f) ? 1.f : 0.f;
        }
        float ctot = block_reduce(cnt, sh.red);
        if (t == 0) counts[m] = (int)(ctot + 0.5f);
        __syncthreads();
    }

    // --- initial cost ---
    {
        float c0 = cost_pass(sh.F, pts4, sh.w, sh.red);
        if (t == 0) sh.cost = c0;
        __syncthreads();
    }

    const int half = lane >> 4, col = lane & 15;

    for (int it = 0; it < LM_ITERS; ++it) {
        float f[9];
#pragma unroll
        for (int i = 0; i < 9; ++i) f[i] = sh.F[i];

        v8f acc = {0.f, 0.f, 0.f, 0.f, 0.f, 0.f, 0.f, 0.f};

        // each wave: stage 128-pt tiles of S = sqrt(w_irls)*[J|r|pad] and WMMA-accumulate
        for (int j = 0; j < TILES_PER_WAVE; ++j) {
            int base = wv * PTS_PER_WAVE + j * TILE;
#pragma unroll
            for (int s4 = 0; s4 < 4; ++s4) {
                int li = s4 * 32 + lane;
                int n = base + li;
                float4 pt = pts4[n];
                float x1 = pt.x, y1 = pt.y, x2 = pt.z, y2 = pt.w;
                float e, jjt;
                epi_terms(f, pt, e, jjt);
                float inv_s = rsqrtf(jjt);
                float r = e * inv_s;
                float wir = sh.w[n] / (1.f + r * r);     // Cauchy IRLS, c2=1
                float swr = sqrtf(wir);
                float si = swr * inv_s;
                float a0 = si * x2, a1 = si * y2, a2 = si;
                float4* dst = (float4*)&sh.S[wv][li * 16];
                dst[0] = make_float4(a0 * x1, a0 * y1, a0, a1 * x1);
                dst[1] = make_float4(a1 * y1, a1, a2 * x1, a2 * y1);
                dst[2] = make_float4(a2, swr * r, 0.f, 0.f);
                dst[3] = make_float4(0.f, 0.f, 0.f, 0.f);
            }
            asm volatile("s_wait_dscnt 0" ::: "memory");
            // M += S_chunk^T * S_chunk : A and B operands share lane layout
            for (int k = 0; k < TILE; k += 4) {
                int row = k + 2 * half;
                v2f ab;
                ab.x = sh.S[wv][row * 16 + col];
                ab.y = sh.S[wv][(row + 1) * 16 + col];
                acc = __builtin_amdgcn_wmma_f32_16x16x4_f32(
                    false, ab, false, ab, (short)0, acc, false, false);
            }
        }

        // dump per-wave 16x16 accumulator, reduce across waves
#pragma unroll
        for (int v = 0; v < 8; ++v)
            sh.Mred[wv][(v + 8 * half) * 16 + col] = acc[v];
        __syncthreads();
        if (t < 100) {
            int i = t / 10, jx = t % 10;
            float s2 = 0.f;
#pragma unroll
            for (int wq = 0; wq < WAVES; ++wq) s2 += sh.Mred[wq][i * 16 + jx];
            sh.M[t] = s2;
        }
        __syncthreads();

        // thread 0: SPD solve (A + lam*I) delta = b, candidate Fn
        if (t == 0) {
            float A[9][9], bb[9], y[9], dx[9];
            float lam = sh.lam;
            for (int i = 0; i < 9; ++i) {
                for (int jx = 0; jx < 9; ++jx) A[i][jx] = sh.M[i * 10 + jx];
                A[i][i] += lam;
                bb[i] = -sh.M[i * 10 + 9];
            }
            // Cholesky (A is SPD: S^T S + lam I)
            for (int k = 0; k < 9; ++k) {
                float d = sqrtf(fmaxf(A[k][k], 1e-30f));
                A[k][k] = d;
                float invd = 1.f / d;
                for (int i = k + 1; i < 9; ++i) A[i][k] *= invd;
                for (int jx = k + 1; jx < 9; ++jx)
                    for (int i = jx; i < 9; ++i) A[i][jx] -= A[i][k] * A[jx][k];
            }
            for (int i = 0; i < 9; ++i) {
                float s = bb[i];
                for (int k = 0; k < i; ++k) s -= A[i][k] * y[k];
                y[i] = s / A[i][i];
            }
            for (int i = 8; i >= 0; --i) {
                float s = y[i];
                for (int k = i + 1; k < 9; ++k) s -= A[k][i] * dx[k];
                dx[i] = s / A[i][i];
            }
            float fn[9], nrm = 0.f;
            for (int i = 0; i < 9; ++i) { fn[i] = sh.F[i] + dx[i]; nrm += fn[i] * fn[i]; }
            float inv = 1.f / (sqrtf(nrm) + EPSF);
            for (int i = 0; i < 9; ++i) sh.Fn[i] = fn[i] * inv;
        }
        __syncthreads();

        float cn = cost_pass(sh.Fn, pts4, sh.w, sh.red);
        if (t == 0) {
            if (cn < sh.cost) {
#pragma unroll
                for (int i = 0; i < 9; ++i) sh.F[i] = sh.Fn[i];
                sh.lam *= 0.1f;
            } else {
                sh.lam *= 10.f;
            }
            sh.cost = fminf(cn, sh.cost);
        }
        __syncthreads();
    }

    if (t < 9) Fout[m * 9 + t] = sh.F[t];
}

__global__ void k_finalize(const float* __restrict__ models,
                           const float* __restrict__ Fn,
                           const int* __restrict__ counts,
                           float* __restrict__ out) {
    __shared__ int ok;
    int t = threadIdx.x;   // 1024 threads, one per model
    if (t == 0) ok = 1;
    __syncthreads();
    if (counts[t] < 8) ok = 0;   // benign race: all writers store 0
    __syncthreads();
    int flag = ok;
#pragma unroll
    for (int k = 0; k < 9; ++k)
        out[t * 9 + k] = flag ? Fn[t * 9 + k] : models[t * 9 + k];
}

extern "C" void kernel_launch(void* const* d_in, const int* in_sizes, int n_in,
                              void* d_out, int out_size, void* d_ws, size_t ws_size,
                              hipStream_t stream) {
    const float* points   = (const float*)d_in[0];  // [4096,4]
    const float* models   = (const float*)d_in[1];  // [1024,3,3]
    const float* prob     = (const float*)d_in[2];  // [4096]
    const float* prob_pow = (const float*)d_in[3];  // [1]
    // d_in[4] = loss_shape (unused in CAUCHY branch)
    float* out = (float*)d_out;

    float* pw     = (float*)d_ws;            // 4096 floats
    float* Fn     = pw + N_PTS;              // 9216 floats
    int*   counts = (int*)(Fn + N_MODELS * 9);  // 1024 ints

    k_pw<<<(N_PTS + 255) / 256, 256, 0, stream>>>(prob, prob_pow, pw);
    k_lm<<<N_MODELS, THREADS, 0, stream>>>(points, models, pw, Fn, counts);
    k_finalize<<<1, N_MODELS, 0, stream>>>(models, Fn, counts, out);
}